// ResidualToAR2_54966991454212
// MI455X (gfx1250) — compile-verified
//
#include <hip/hip_runtime.h>
#include <math.h>
#include <stdint.h>

typedef __attribute__((ext_vector_type(2))) float v2f;
typedef __attribute__((ext_vector_type(8))) float v8f;
typedef __attribute__((ext_vector_type(4))) unsigned int v4u;
typedef __attribute__((ext_vector_type(8))) int v8i;
typedef __attribute__((ext_vector_type(4))) int v4i;

#define LFULL 2048
#define HOUT  96
#define NROWS 16384          // 32*512
#define LAMR  0.001f
#define CLIPK 4.0f

// ---------------------------------------------------------------------------
// Kernel 1: one wave32 per row (unchanged from round 1).
// Also warms L2 (192MB) with the full 134MB x tensor for the GEMM kernel.
// ---------------------------------------------------------------------------
__global__ __launch_bounds__(256) void ar2_stats_kernel(
    const float* __restrict__ x, float* __restrict__ yb_out,
    float* __restrict__ lohi) {
  const int wave = threadIdx.x >> 5;
  const int lane = threadIdx.x & 31;
  const int row  = blockIdx.x * 8 + wave;
  const float* __restrict__ xr = x + (size_t)row * LFULL;

  float s = 0.f, ss = 0.f;
  for (int k = lane; k < LFULL; k += 32) {
    float v = xr[k];
    s += v; ss += v * v;
  }

  const float* __restrict__ y = xr + (LFULL - 150);
  float p11 = 0.f, p22 = 0.f, p12 = 0.f, b1 = 0.f, b2 = 0.f;
  for (int j = lane; j < 148; j += 32) {
    float y0 = y[j], y1 = y[j + 1], y2v = y[j + 2];
    p11 += y1 * y1;
    p22 += y0 * y0;
    p12 += y1 * y0;
    b1  += y1 * y2v;
    b2  += y0 * y2v;
  }

  for (int m = 16; m >= 1; m >>= 1) {
    s   += __shfl_xor(s,   m, 32);
    ss  += __shfl_xor(ss,  m, 32);
    p11 += __shfl_xor(p11, m, 32);
    p22 += __shfl_xor(p22, m, 32);
    p12 += __shfl_xor(p12, m, 32);
    b1  += __shfl_xor(b1,  m, 32);
    b2  += __shfl_xor(b2,  m, 32);
  }

  if (lane == 0) {
    float A11 = p11 + LAMR, A22 = p22 + LAMR, A12 = p12;
    float det = A11 * A22 - A12 * A12;
    float a1 = (b1 * A22 - b2 * A12) / det;
    float a2 = (A11 * b2 - A12 * b1) / det;

    float mean = s * (1.0f / (float)LFULL);
    float var  = (ss - (float)LFULL * mean * mean) * (1.0f / (float)(LFULL - 1));
    var = fmaxf(var, 0.f);
    float sd = fmaxf(sqrtf(var), 1e-6f);
    float last = xr[LFULL - 1];
    lohi[2 * row + 0] = last - CLIPK * sd;
    lohi[2 * row + 1] = last + CLIPK * sd;

    float y1 = xr[LFULL - 1];
    float y2v = xr[LFULL - 2];
    float* __restrict__ o = yb_out + (size_t)row * HOUT;
    for (int h = 0; h < HOUT; ++h) {
      float yn = a1 * y1 + a2 * y2v;
      o[h] = yn;
      y2v = y1;
      y1 = yn;
    }
  }
}

// ---------------------------------------------------------------------------
// Kernel 2: fp32 WMMA GEMM with TDM-staged, double-buffered LDS.
// Block: 128 threads = 4 waves; tile M=64, N=96 (6 subtiles/wave), K chunks 32.
// X tile (64x32 f32, 8KB) is DMA'd by the Tensor Data Mover directly into LDS
// with hardware padding (32 dwords + 4 pad -> row stride 36 = conflict-free).
// W chunk is staged transposed by ALU (TDM cannot transpose), double-buffered.
// ---------------------------------------------------------------------------
#define KC  32
#define LDP 36                 // padded LDS row stride in floats
#define NK  (LFULL / KC)       // 64 K-chunks

// Issue one TDM tensor load: 64 rows x KC cols f32 tile of x -> LDS (padded).
// Built per CDNA5 ISA D# layout; inline asm form is portable
// across both toolchains' differing builtin arities.
__device__ __forceinline__ void tdm_load_x_tile(const float* x, int row0,
                                                int k0, float* ldsdst) {
  unsigned lds_addr = (unsigned)(uintptr_t)ldsdst;  // low 32 bits = LDS offset
  unsigned long long ga =
      (unsigned long long)(uintptr_t)(x + (size_t)row0 * LFULL + k0);

  v4u g0;
  g0.x = 1u;                                   // count=1, is_restore=0
  g0.y = lds_addr;                             // lds_addr[31:0]
  g0.z = (unsigned)(ga & 0xffffffffu);         // global_addr[31:0]
  g0.w = (unsigned)((ga >> 32) & 0x01ffffffu)  // global_addr[56:32]
         | (2u << 30);                         // type=2 ("image")

  v8i g1;
  // data_size=2 (4B), pad_enable=1, pad_interval=4 (32 dwords), pad_amount=3 (4 dwords)
  g1[0] = (2 << 16) | (1 << 20) | (4 << 22) | (3 << 25);
  g1[1] = (int)(((unsigned)LFULL & 0xffffu) << 16);          // tensor_dim0 lo16
  g1[2] = (int)((((unsigned)LFULL >> 16) & 0xffffu)          // tensor_dim0 hi16
                | (((unsigned)NROWS & 0xffffu) << 16));      // tensor_dim1 lo16
  g1[3] = (int)((((unsigned)NROWS >> 16) & 0xffffu)          // tensor_dim1 hi16
                | ((unsigned)KC << 16));                     // tile_dim0 = KC
  g1[4] = 64;                                                // tile_dim1=64, tile_dim2=0
  g1[5] = LFULL;                                             // tensor_dim0_stride lo32
  g1[6] = 0;                                                 // stride hi16, dim1_stride lo
  g1[7] = 0;

  v4i g2 = {0, 0, 0, 0};
  v4i g3 = {0, 0, 0, 0};

  asm volatile("tensor_load_to_lds %0, %1, %2, %3"
               :: "s"(g0), "s"(g1), "s"(g2), "s"(g3)
               : "memory");
}

__global__ __launch_bounds__(128) void wmma_gemm_clip_kernel(
    const float* __restrict__ x, const float* __restrict__ w,
    float* __restrict__ out, const float* __restrict__ lohi) {
  __shared__ float Xs[2][64 * LDP];   // 2 x 9216 B  (TDM destination)
  __shared__ float Wt[2][96 * LDP];   // 2 x 13824 B (W chunk, transposed)

  const int t    = threadIdx.x;
  const int wave = t >> 5;
  const int lane = t & 31;
  const int ln   = lane & 15;
  const int koff = (lane < 16) ? 0 : 2;   // A/B fragment K offset per half-wave
  const int row0 = blockIdx.x * 64;

  v8f acc[6];
#pragma unroll
  for (int n = 0; n < 6; ++n) acc[n] = 0.0f;

  // ---- prologue: stage chunk 0 ----
  if (wave == 0) tdm_load_x_tile(x, row0, 0, &Xs[0][0]);   // TENSORcnt=1 (wave0)
  for (int j = t; j < KC * HOUT; j += 128) {
    int kk = j / HOUT, h = j - kk * HOUT;
    Wt[0][h * LDP + kk] = w[(size_t)kk * HOUT + h];
  }

  for (int i = 0; i < NK; ++i) {
    const int cur = i & 1;
    const int nxt = cur ^ 1;

    // ---- stage chunk i+1 into 'nxt' while chunk i finishes / is computed ----
    if (i + 1 < NK) {
      const int k1 = (i + 1) * KC;
      if (wave == 0) tdm_load_x_tile(x, row0, k1, &Xs[nxt][0]);
      for (int j = t; j < KC * HOUT; j += 128) {
        int kk = j / HOUT, h = j - kk * HOUT;
        Wt[nxt][h * LDP + kk] = w[(size_t)(k1 + kk) * HOUT + h];
      }
    }

    // chunk i's TDM (issued last iteration) must have landed; chunk i+1 may
    // still be in flight (TENSORcnt in-order => wait <=1 proves oldest done).
    if (wave == 0) {
      if (i + 1 < NK) __builtin_amdgcn_s_wait_tensorcnt(1);
      else            __builtin_amdgcn_s_wait_tensorcnt(0);
    }
    __syncthreads();   // publish Xs[cur] (TDM) + Wt[cur] to all waves

    // ---- WMMA on chunk i ----
#pragma unroll
    for (int kk = 0; kk < KC; kk += 4) {
      v2f a = *(const v2f*)(&Xs[cur][(wave * 16 + ln) * LDP + kk + koff]);
#pragma unroll
      for (int n = 0; n < 6; ++n) {
        v2f b = *(const v2f*)(&Wt[cur][(n * 16 + ln) * LDP + kk + koff]);
        acc[n] = __builtin_amdgcn_wmma_f32_16x16x4_f32(
            false, a, false, b, (short)0, acc[n], false, false);
      }
    }
    __syncthreads();   // all waves done reading 'cur' before it is re-staged
  }

  // epilogue: C layout (16x16 f32): VGPR v -> lanes 0-15: M=v, lanes 16-31: M=v+8
  const int half = lane >> 4;
#pragma unroll
  for (int n = 0; n < 6; ++n) {
#pragma unroll
    for (int v = 0; v < 8; ++v) {
      int row = row0 + wave * 16 + v + 8 * half;
      int h   = n * 16 + ln;
      float lo = lohi[2 * row + 0];
      float hi = lohi[2 * row + 1];
      size_t idx = (size_t)row * HOUT + h;
      float r = out[idx] + acc[n][v];   // out currently holds Yb
      r = fminf(fmaxf(r, lo), hi);
      out[idx] = r;
    }
  }
}

extern "C" void kernel_launch(void* const* d_in, const int* in_sizes, int n_in,
                              void* d_out, int out_size, void* d_ws, size_t ws_size,
                              hipStream_t stream) {
  const float* x = (const float*)d_in[0];   // (32, 512, 2048) f32
  const float* w = (const float*)d_in[1];   // (2048, 96) f32
  float* out  = (float*)d_out;              // (32, 512, 96) f32
  float* lohi = (float*)d_ws;               // 2 floats per row = 128 KB

  ar2_stats_kernel<<<NROWS / 8, 256, 0, stream>>>(x, out, lohi);
  wmma_gemm_clip_kernel<<<NROWS / 64, 128, 0, stream>>>(x, w, out, lohi);
}